// MultiBoxLoss_offset_54271206752707
// MI455X (gfx1250) — compile-verified
//
#include <hip/hip_runtime.h>
#include <hip/hip_bf16.h>

typedef __attribute__((ext_vector_type(2))) float v2f;
typedef __attribute__((ext_vector_type(8))) float v8f;

#define NUM_B 64
#define NUM_P 24564
#define NUM_CL 21
#define BP (NUM_B * NUM_P)
#define NEG_INF (-3.0e38f)
#define EQCAP 512

// ---------- helpers ----------

__device__ __forceinline__ float sl1(float d) {
  d = fabsf(d);
  return (d < 1.0f) ? 0.5f * d * d : d - 0.5f;
}

// block-wide sum; result valid in thread 0. `s` must hold >= blockDim.x/32 floats.
__device__ __forceinline__ float block_sum(float v, float* s) {
  const int lane = threadIdx.x & 31;
  const int w = threadIdx.x >> 5;
#pragma unroll
  for (int o = 16; o > 0; o >>= 1) v += __shfl_down(v, o, 32);
  if (lane == 0) s[w] = v;
  __syncthreads();
  const int nw = blockDim.x >> 5;
  float r = (threadIdx.x < nw) ? s[threadIdx.x] : 0.0f;
  if (w == 0) {
#pragma unroll
    for (int o = 16; o > 0; o >>= 1) r += __shfl_down(r, o, 32);
  }
  __syncthreads();  // allow scratch reuse
  return r;
}

// ---------- Kernel A: 21-class CE via WMMA f32 16x16x4 sum-of-exp ----------
// grid = (24, 64), block = 256 (8 waves). Each wave: 8 iterations x 16 priors.
__global__ void __launch_bounds__(256) ce_wmma_kernel(
    const float* __restrict__ conf, const int* __restrict__ conf_t,
    float* __restrict__ cemine, float* __restrict__ acc,
    float* __restrict__ numpos_row) {
  __shared__ float red[8];
  const int lane = threadIdx.x & 31;
  const int wid = threadIdx.x >> 5;
  const int half = lane >> 4;   // 0: K=0,1  1: K=2,3  (ISA A-matrix layout)
  const int off = half * 2;
  const int m = lane & 15;      // A-matrix row (prior within tile)
  const int b = blockIdx.y;
  const int waveBase = blockIdx.x * 1024 + wid * 128;

  float accCe = 0.0f, accNp = 0.0f;

  for (int it = 0; it < 8; ++it) {
    const int prior = waveBase + it * 16 + m;
    const bool valid = prior < NUM_P;
    const int pc = valid ? prior : (NUM_P - 1);
    const long base = ((long)b * NUM_P + pc) * NUM_CL;

    // 12 class slots per lane: classes 4t+off, 4t+off+1 for t=0..5 (pad >=21)
    float x[12];
#pragma unroll
    for (int t = 0; t < 6; ++t) {
      const int c0 = 4 * t + off;
      x[2 * t]     = (c0 < NUM_CL)     ? conf[base + c0]     : NEG_INF;
      x[2 * t + 1] = (c0 + 1 < NUM_CL) ? conf[base + c0 + 1] : NEG_INF;
    }

    // full per-prior max: local 12 + partner half via xor-16
    float mx = x[0];
#pragma unroll
    for (int i = 1; i < 12; ++i) mx = fmaxf(mx, x[i]);
    mx = fmaxf(mx, __shfl_xor(mx, 16, 32));

    // gather logits[label]
    const int label = conf_t[(long)b * NUM_P + pc];
    float g = NEG_INF;
#pragma unroll
    for (int t = 0; t < 6; ++t) {
#pragma unroll
      for (int j = 0; j < 2; ++j) {
        const int c = 4 * t + off + j;
        g = (c == label) ? x[2 * t + j] : g;
      }
    }
    g = fmaxf(g, __shfl_xor(g, 16, 32));

    // exp(x - max); pad slots -> 0
    float e[12];
#pragma unroll
    for (int i = 0; i < 12; ++i)
      e[i] = (x[i] > -1.0e37f) ? __expf(x[i] - mx) : 0.0f;

    // sum over 24 (21 valid) class slots: 6 chained WMMAs, B = ones.
    // D[m][n] = sum_k exp; every column of row m equals sumexp(prior m).
    v8f c = {};
    const v2f ones = {1.0f, 1.0f};
#pragma unroll
    for (int t = 0; t < 6; ++t) {
      v2f a = {e[2 * t], e[2 * t + 1]};
      c = __builtin_amdgcn_wmma_f32_16x16x4_f32(false, a, false, ones,
                                                (short)0, c, false, false);
    }

    // extract sumexp for prior (= lane) into lanes 0..15:
    // lanes 0-15 hold rows 0-7 (VGPR j = M=j); lanes 16-31 hold rows 8-15.
    const int i3 = lane & 7;
    float v = c[0];
    v = (i3 == 1) ? c[1] : v;
    v = (i3 == 2) ? c[2] : v;
    v = (i3 == 3) ? c[3] : v;
    v = (i3 == 4) ? c[4] : v;
    v = (i3 == 5) ? c[5] : v;
    v = (i3 == 6) ? c[6] : v;
    v = (i3 == 7) ? c[7] : v;
    const float vOther = __shfl_xor(v, 16, 32);
    const float sumexp = (lane & 8) ? vOther : v;

    if (lane < 16 && valid) {
      const float ce = __logf(sumexp) + mx - g;  // lse - logits[label]
      const bool pos = label > 0;
      cemine[(long)b * NUM_P + prior] = pos ? 0.0f : ce;
      if (pos) { accCe += ce; accNp += 1.0f; }
    }
  }

  const float s1 = block_sum(accCe, red);
  const float s2 = block_sum(accNp, red);
  if (threadIdx.x == 0) {
    atomicAdd(&acc[3], s1);          // sum_pos(ce)
    atomicAdd(&numpos_row[b], s2);   // per-row num_pos
    atomicAdd(&acc[7], s2);          // total num_pos (N)
  }
}

// ---------- Kernel B: smooth-L1 losses + 2-class has_lp CE ----------
__global__ void __launch_bounds__(256) elem_kernel(
    const float* __restrict__ loc_data, const float* __restrict__ has_lp_data,
    const float* __restrict__ size_lp_data, const float* __restrict__ offset_data,
    const float* __restrict__ loc_t, const int* __restrict__ conf_t,
    const int* __restrict__ has_lp_t, const float* __restrict__ size_lp_t,
    const float* __restrict__ offset_t, float* __restrict__ hlpce,
    float* __restrict__ acc) {
  __shared__ float red[8];
  const long idx = (long)blockIdx.x * blockDim.x + threadIdx.x;  // BP % 256 == 0

  // CDNA5 prefetch hint (global_prefetch_b8) on the dominant stream
  __builtin_prefetch(loc_data + idx * 4 + 4096, 0, 0);

  const int label = conf_t[idx];
  const float posf = (label > 0) ? 1.0f : 0.0f;
  const int ht = has_lp_t[idx];
  const float hlpf = (float)ht;

  float al = 0.0f;
#pragma unroll
  for (int j = 0; j < 4; ++j)
    al += sl1(loc_data[idx * 4 + j] - loc_t[idx * 4 + j]);
  al *= posf;

  float as = sl1(size_lp_data[idx * 2] - size_lp_t[idx * 2]) +
             sl1(size_lp_data[idx * 2 + 1] - size_lp_t[idx * 2 + 1]);
  as *= posf * hlpf;  // (p*h - t*h) == h*(p-t), h in {0,1}

  float ao = sl1(offset_data[idx * 2] - offset_t[idx * 2]) +
             sl1(offset_data[idx * 2 + 1] - offset_t[idx * 2 + 1]);
  ao *= posf * hlpf;

  const float h0 = has_lp_data[idx * 2];
  const float h1 = has_lp_data[idx * 2 + 1];
  const float hm = fmaxf(h0, h1);
  const float lse = hm + __logf(__expf(h0 - hm) + __expf(h1 - hm));
  const float ceh = lse - (ht ? h1 : h0);
  hlpce[idx] = ceh;
  const float ah = posf * ceh;

  float s;
  s = block_sum(al, red); if (threadIdx.x == 0) atomicAdd(&acc[0], s);
  s = block_sum(as, red); if (threadIdx.x == 0) atomicAdd(&acc[1], s);
  s = block_sum(ao, red); if (threadIdx.x == 0) atomicAdd(&acc[2], s);
  s = block_sum(ah, red); if (threadIdx.x == 0) atomicAdd(&acc[4], s);
}

// ---------- Kernel C: per-row exact top-k radix select (hard neg mining) ----------
// One block per batch row. 3 LDS-histogram passes (11/11/10 bits of the
// non-negative float bit pattern), then sum strict-greater + stable tie handling.
__global__ void __launch_bounds__(256) select_kernel(
    const float* __restrict__ cemine, const float* __restrict__ hlpce,
    const float* __restrict__ numpos_row, float* __restrict__ acc) {
  __shared__ int hist[2048];
  __shared__ int s_bin, s_rem;
  __shared__ int eqlist[EQCAP];
  __shared__ int eqcount;
  __shared__ float red[8];

  const int b = blockIdx.x;
  const float* row = cemine + (long)b * NUM_P;
  const float* hrow = hlpce + (long)b * NUM_P;
  const int np = (int)(numpos_row[b] + 0.5f);
  int k = 3 * np;
  if (k > NUM_P - 1) k = NUM_P - 1;
  // pos entries have ce_mine == 0.0 and all non-pos > 0, so the effective
  // negative set is the top min(k, #non-pos) of the strictly-positive values.
  int keff = (k < NUM_P - np) ? k : (NUM_P - np);
  if (keff <= 0) return;  // uniform per block
  const int tid = threadIdx.x;

  // pass 1: bits [31:21]
  for (int i = tid; i < 2048; i += blockDim.x) hist[i] = 0;
  __syncthreads();
  for (int p = tid; p < NUM_P; p += blockDim.x)
    atomicAdd(&hist[__float_as_uint(row[p]) >> 21], 1);
  __syncthreads();
  if (tid == 0) {
    int rem = keff, bin = 2047;
    for (; bin > 0; --bin) { int c = hist[bin]; if (rem <= c) break; rem -= c; }
    s_bin = bin; s_rem = rem;
  }
  __syncthreads();
  const unsigned bin1 = (unsigned)s_bin;

  // pass 2: bits [20:10] within bin1
  for (int i = tid; i < 2048; i += blockDim.x) hist[i] = 0;
  __syncthreads();
  for (int p = tid; p < NUM_P; p += blockDim.x) {
    const unsigned u = __float_as_uint(row[p]);
    if ((u >> 21) == bin1) atomicAdd(&hist[(u >> 10) & 0x7FFu], 1);
  }
  __syncthreads();
  if (tid == 0) {
    int rem = s_rem, bin = 2047;
    for (; bin > 0; --bin) { int c = hist[bin]; if (rem <= c) break; rem -= c; }
    s_bin = bin; s_rem = rem;
  }
  __syncthreads();
  const unsigned bin2 = (unsigned)s_bin;
  const unsigned pre2 = (bin1 << 11) | bin2;

  // pass 3: bits [9:0] within (bin1,bin2)
  for (int i = tid; i < 1024; i += blockDim.x) hist[i] = 0;
  __syncthreads();
  for (int p = tid; p < NUM_P; p += blockDim.x) {
    const unsigned u = __float_as_uint(row[p]);
    if ((u >> 10) == pre2) atomicAdd(&hist[u & 0x3FFu], 1);
  }
  __syncthreads();
  if (tid == 0) {
    int rem = s_rem, bin = 1023;
    for (; bin > 0; --bin) { int c = hist[bin]; if (rem <= c) break; rem -= c; }
    s_bin = bin; s_rem = rem;
  }
  __syncthreads();
  const unsigned T = (bin1 << 21) | (bin2 << 10) | (unsigned)s_bin;
  const int need_eq = s_rem;  // how many threshold-equal elements to take (by index order)

  if (tid == 0) eqcount = 0;
  __syncthreads();

  // pass 4: sum strict-greater; collect threshold-equal indices
  float sc = 0.0f, sh = 0.0f;
  for (int p = tid; p < NUM_P; p += blockDim.x) {
    const unsigned u = __float_as_uint(row[p]);
    if (u > T) {
      sc += row[p];
      sh += hrow[p];
    } else if (u == T) {
      const int i = atomicAdd(&eqcount, 1);
      if (i < EQCAP) eqlist[i] = p;
    }
  }
  float ssc = block_sum(sc, red);
  float ssh = block_sum(sh, red);
  __syncthreads();
  if (tid == 0) {
    int n = eqcount; if (n > EQCAP) n = EQCAP;
    for (int i = 1; i < n; ++i) {  // sort indices ascending (argsort stability)
      const int key = eqlist[i];
      int j = i - 1;
      while (j >= 0 && eqlist[j] > key) { eqlist[j + 1] = eqlist[j]; --j; }
      eqlist[j + 1] = key;
    }
    const int take = (need_eq < n) ? need_eq : n;
    for (int i = 0; i < take; ++i) { ssc += row[eqlist[i]]; ssh += hrow[eqlist[i]]; }
    atomicAdd(&acc[5], ssc);  // sum_neg(ce)
    atomicAdd(&acc[6], ssh);  // sum_neg(ce_has_lp)
  }
}

// ---------- Kernel D: finalize 5 scalars ----------
__global__ void finalize_kernel(const float* __restrict__ acc,
                                float* __restrict__ out) {
  if (threadIdx.x == 0 && blockIdx.x == 0) {
    const float N = acc[7];
    out[0] = acc[0] / N;               // loss_l
    out[1] = (acc[3] + acc[5]) / N;    // loss_c
    out[2] = acc[1] / N;               // loss_size_lp
    out[3] = acc[2] / N;               // loss_offset
    out[4] = (acc[4] + acc[6]) / N;    // loss_has_lp
  }
}

// ---------- launch ----------
extern "C" void kernel_launch(void* const* d_in, const int* in_sizes, int n_in,
                              void* d_out, int out_size, void* d_ws,
                              size_t ws_size, hipStream_t stream) {
  const float* loc_data     = (const float*)d_in[0];
  const float* conf_data    = (const float*)d_in[1];
  const float* has_lp_data  = (const float*)d_in[2];
  const float* size_lp_data = (const float*)d_in[3];
  const float* offset_data  = (const float*)d_in[4];
  const float* loc_t        = (const float*)d_in[5];
  const int*   conf_t       = (const int*)d_in[6];
  const int*   has_lp_t     = (const int*)d_in[7];
  const float* size_lp_t    = (const float*)d_in[8];
  const float* offset_t     = (const float*)d_in[9];

  float* ws = (float*)d_ws;
  float* cemine     = ws;                       // [B*P]
  float* hlpce      = ws + (size_t)BP;          // [B*P]
  float* numpos_row = ws + 2 * (size_t)BP;      // [64]
  float* acc        = numpos_row + NUM_B;       // [8]

  hipMemsetAsync(numpos_row, 0, (NUM_B + 8) * sizeof(float), stream);

  dim3 gA(24, NUM_B);  // 24*1024 >= 24564 priors per row
  ce_wmma_kernel<<<gA, 256, 0, stream>>>(conf_data, conf_t, cemine, acc,
                                         numpos_row);
  elem_kernel<<<BP / 256, 256, 0, stream>>>(
      loc_data, has_lp_data, size_lp_data, offset_data, loc_t, conf_t,
      has_lp_t, size_lp_t, offset_t, hlpce, acc);
  select_kernel<<<NUM_B, 256, 0, stream>>>(cemine, hlpce, numpos_row, acc);
  finalize_kernel<<<1, 32, 0, stream>>>(acc, (float*)d_out);
}